// ChamferLossSqrtT_76175539962301
// MI455X (gfx1250) — compile-verified
//
#include <hip/hip_runtime.h>
#include <float.h>
#include <math.h>

typedef __attribute__((ext_vector_type(2))) float v2f;
typedef __attribute__((ext_vector_type(8))) float v8f;

#define NPTS 4096
#define BS   32
#define RT   2      // row tiles (A fragments) per wave

__global__ void chamfer_zero_ws(float* ws) {
  ws[threadIdx.x] = 0.0f;
}

// One wave holds RT 16-row A tiles and streams all 4096 B points through
// V_WMMA_F32_16X16X4_F32 (K padded 3->4 with zero).
// D[m][n] = ||b_n||^2 - 2 a_m.b_n  (C seeded with ||b_n||^2, B scaled by -2);
// ||a_m||^2 is row-constant and added after the running min (min/max commute).
__global__ __launch_bounds__(256) void chamfer_min_kernel(
    const float* __restrict__ points, const float* __restrict__ gts,
    float* __restrict__ acc) {
  const int b   = blockIdx.y;
  const int dir = blockIdx.z;                 // 0: points->gts, 1: gts->points
  const float* A = (dir == 0) ? points : gts;
  const float* B = (dir == 0) ? gts : points;

  const int lane = threadIdx.x & 31;
  const int wave = threadIdx.x >> 5;
  const int half = lane >> 4;                 // 0: k={0,1}=(x,y), 1: k={2,3}=(z,0)
  const bool hi  = (half != 0);
  const int r    = lane & 15;                 // row (A) / column (B) index in tile
  const int m0   = (blockIdx.x * 8 + wave) * (16 * RT);

  // ---- A fragments: 16x4 f32 each, all loads unconditional (branchless) ----
  v2f   afrag[RT];
  float a2_row[RT][8];
  float minv[RT][8];
  #pragma unroll
  for (int t = 0; t < RT; ++t) {
    const float* Ap = A + ((size_t)b * NPTS + (size_t)(m0 + t * 16 + r)) * 3;
    float ax = Ap[0], ay = Ap[1], az = Ap[2];
    float a2 = ax * ax + ay * ay + az * az;   // full norm in every lane
    afrag[t].x = hi ? az : ax;
    afrag[t].y = hi ? 0.0f : ay;
    #pragma unroll
    for (int v = 0; v < 8; ++v)               // line a2 up with D layout rows
      a2_row[t][v] = __shfl(a2, v + (half << 3), 32);
    #pragma unroll
    for (int v = 0; v < 8; ++v) minv[t][v] = FLT_MAX;
  }

  const float* Bb = B + (size_t)b * NPTS * 3;
  for (int nt = 0; nt < NPTS / 16; ++nt) {
    const float* Bp = Bb + (size_t)(nt * 16 + r) * 3;
    float bx = Bp[0], by = Bp[1], bz = Bp[2]; // unconditional loads
    float b2 = bx * bx + by * by + bz * bz;   // ||b_col||^2, no shuffle needed

    v2f bfrag;
    bfrag.x = -2.0f * (hi ? bz : bx);         // v_cndmask + mul, branchless
    bfrag.y = hi ? 0.0f : (-2.0f * by);

    v8f c;
    #pragma unroll
    for (int v = 0; v < 8; ++v) c[v] = b2;    // column-uniform seed

    #pragma unroll
    for (int t = 0; t < RT; ++t) {
      v8f d = __builtin_amdgcn_wmma_f32_16x16x4_f32(
          false, afrag[t], false, bfrag, (short)0, c, false, false);
      #pragma unroll
      for (int v = 0; v < 8; ++v) minv[t][v] = fminf(minv[t][v], d[v]);
    }
  }

  // Min over 16 columns (xor masks 1,2,4,8 stay inside each 16-lane half),
  // then add ||a||^2, clamp, sqrt, and accumulate.
  float sum = 0.0f;
  #pragma unroll
  for (int t = 0; t < RT; ++t) {
    #pragma unroll
    for (int v = 0; v < 8; ++v) {
      float m = minv[t][v];
      m = fminf(m, __shfl_xor(m, 1, 32));
      m = fminf(m, __shfl_xor(m, 2, 32));
      m = fminf(m, __shfl_xor(m, 4, 32));
      m = fminf(m, __shfl_xor(m, 8, 32));
      sum += sqrtf(fmaxf(a2_row[t][v] + m, 0.0f));
    }
  }

  if (r == 0)   // lane 0 covers rows v, lane 16 covers rows v+8 (per tile)
    atomicAdd(&acc[dir * BS + b], sum);
}

__global__ void chamfer_finalize(const float* __restrict__ acc,
                                 float* __restrict__ out) {
  int b = threadIdx.x;                        // 0..31 = batch index
  float p2g  = acc[b]      * (1.0f / NPTS);
  float g2p  = acc[BS + b] * (1.0f / NPTS);
  float loss = p2g + g2p;
  #pragma unroll
  for (int m = 16; m >= 1; m >>= 1) {
    p2g  += __shfl_xor(p2g,  m, 32);
    g2p  += __shfl_xor(g2p,  m, 32);
    loss += __shfl_xor(loss, m, 32);
  }
  if (b == 0) {
    out[0] = loss * (1.0f / BS);
    out[1] = p2g  * (1.0f / BS);
    out[2] = g2p  * (1.0f / BS);
  }
}

extern "C" void kernel_launch(void* const* d_in, const int* in_sizes, int n_in,
                              void* d_out, int out_size, void* d_ws, size_t ws_size,
                              hipStream_t stream) {
  const float* points = (const float*)d_in[0];
  const float* gts    = (const float*)d_in[1];
  float* acc = (float*)d_ws;                  // 64 floats: [dir][batch]

  chamfer_zero_ws<<<1, 64, 0, stream>>>(acc);
  dim3 grid(NPTS / (16 * RT * 8), BS, 2);
  chamfer_min_kernel<<<grid, 256, 0, stream>>>(points, gts, acc);
  chamfer_finalize<<<1, 32, 0, stream>>>(acc, (float*)d_out);
}